// LTM_transfer_20856361190031
// MI455X (gfx1250) — compile-verified
//
#include <hip/hip_runtime.h>
#include <hip/hip_bf16.h>

// Problem constants (from the reference)
#define HSZ 256
#define WSZ 256
#define RNUM 81          // 9x9 displacement taps, du fastest
#define HALO 4

// Tiling: 128 threads (4 waves), tile = 128 wide x 4 tall, 4 px/thread in x.
#define TILE_W 128
#define TILE_H 4
#define NTHREADS 128
#define LW (TILE_W + 2 * HALO)   // 136
#define LH (TILE_H + 2 * HALO)   // 12

typedef __attribute__((ext_vector_type(4))) float float4v;
typedef __attribute__((address_space(1))) int* gbl_i32p;
typedef __attribute__((address_space(3))) int* lds_i32p;

#if defined(__gfx1250__) && __has_builtin(__builtin_amdgcn_global_load_async_to_lds_b32)
#define USE_ASYNC_LDS 1
#else
#define USE_ASYNC_LDS 0
#endif

__global__ __launch_bounds__(NTHREADS, 1) void ltm_transfer_kernel(
    const float* __restrict__ sim,   // (N, 81, 256, 256)
    const float* __restrict__ fmap,  // (N, 1, 256, 256)
    float* __restrict__ out)         // (N, 1, 256, 256)
{
  __shared__ float tile[LH * LW];    // zero-padded f_map halo tile

  const int tid = (int)threadIdx.x;
  const int n   = (int)blockIdx.z;
  const int x0  = (int)blockIdx.x * TILE_W;
  const int y0  = (int)blockIdx.y * TILE_H;

  // ---- Stage f_map halo tile into LDS: async loads for in-bounds, 0 for OOB.
  // Zero-padding here removes ALL masking work from the 81-tap hot loop.
  const float* fbase = fmap + (size_t)n * HSZ * WSZ;
  for (int i = tid; i < LH * LW; i += NTHREADS) {
    const int row = i / LW;
    const int col = i - row * LW;
    const int gy  = y0 - HALO + row;
    const int gx  = x0 - HALO + col;
    const bool inb = ((unsigned)gy < (unsigned)HSZ) && ((unsigned)gx < (unsigned)WSZ);
    if (inb) {
      const float* src = fbase + gy * WSZ + gx;
#if USE_ASYNC_LDS
      __builtin_amdgcn_global_load_async_to_lds_b32(
          (gbl_i32p)(void*)src,
          (lds_i32p)(void*)&tile[i],
          /*imm offset*/ 0, /*cpol*/ 0);
#else
      tile[i] = *src;
#endif
    } else {
      tile[i] = 0.0f;
    }
  }
#if defined(__gfx1250__) && __has_builtin(__builtin_amdgcn_s_wait_asynccnt)
  __builtin_amdgcn_s_wait_asynccnt(0);
#elif defined(__gfx1250__)
  asm volatile("s_wait_asynccnt 0" ::: "memory");
#endif
  __syncthreads();

  // ---- Per-thread pixel group: 4 consecutive x.
  const int ty  = tid >> 5;          // 0..3 (one wave per row)
  const int txq = (tid & 31) << 2;   // 0,4,...,124
  const int y   = y0 + ty;
  const int x   = x0 + txq;

  // sim stream: one b128 per tap per thread; r-stride = H*W floats. Read once
  // -> non-temporal so the 170MB stream doesn't evict anything useful.
  const float* sp = sim + (((size_t)n * RNUM) * HSZ + (size_t)y) * WSZ + x;

  const float LOG2E = 1.44269504088896340736f;

  float m[4], dsum[4], acc[4];
#pragma unroll
  for (int p = 0; p < 4; ++p) { m[p] = -1e30f; dsum[p] = 0.0f; acc[p] = 0.0f; }

#pragma unroll 1
  for (int dv = 0; dv < 9; ++dv) {
    // 12-float LDS row window covers all 9 du taps for 4 pixels.
    // (ty+dv)*LW*4 and txq*4 are both 16B-multiples -> aligned ds_load_b128.
    const float4v* lrow = (const float4v*)&tile[(ty + dv) * LW + txq];
    float w[12];
    *(float4v*)&w[0] = lrow[0];
    *(float4v*)&w[4] = lrow[1];
    *(float4v*)&w[8] = lrow[2];

#pragma unroll
    for (int du = 0; du < 9; ++du) {
      const float4v v = __builtin_nontemporal_load((const float4v*)sp);
      sp += (size_t)HSZ * WSZ;
#pragma unroll
      for (int p = 0; p < 4; ++p) {
        // Online softmax, 1 transcendental per element:
        // exactly one of {rescale, new term} equals exp2(-|vv-m|), other is 1.
        const float vv = v[p] * LOG2E;        // log2 domain
        const float mo = m[p];
        const float u  = __builtin_amdgcn_exp2f(-__builtin_fabsf(vv - mo));
        const bool up  = vv > mo;
        const float s  = up ? u : 1.0f;       // scale for old accumulators
        const float e  = up ? 1.0f : u;       // weight of new term
        m[p]    = up ? vv : mo;
        dsum[p] = dsum[p] * s + e;
        acc[p]  = acc[p] * s + e * w[p + du];
      }
    }
  }

  float4v o;
#pragma unroll
  for (int p = 0; p < 4; ++p) o[p] = acc[p] / dsum[p];
  __builtin_nontemporal_store(
      o, (float4v*)(out + ((size_t)n * HSZ + (size_t)y) * WSZ + x));
}

extern "C" void kernel_launch(void* const* d_in, const int* in_sizes, int n_in,
                              void* d_out, int out_size, void* d_ws, size_t ws_size,
                              hipStream_t stream) {
  const float* sim  = (const float*)d_in[0];  // (N,81,256,256) f32
  const float* fmap = (const float*)d_in[1];  // (N,1,256,256)  f32
  float* out        = (float*)d_out;          // (N,1,256,256)  f32

  const int N = in_sizes[1] / (HSZ * WSZ);    // 8
  dim3 grid(WSZ / TILE_W, HSZ / TILE_H, N);   // (2, 64, 8) -> 1024 blocks
  ltm_transfer_kernel<<<grid, NTHREADS, 0, stream>>>(sim, fmap, out);
}